// Recon_loss_6708738916982
// MI455X (gfx1250) — compile-verified
//
#include <hip/hip_runtime.h>
#include <math.h>

// CDNA5 (gfx1250) wave32 WMMA chamfer-loss kernel, round 2.
//
// Squared distance as one V_WMMA_F32_16X16X4_F32 per 16x16 tile:
//   A(16x4) = [-2ax, -2ay, -2az, 1 ]   (per-row query point)
//   B(4x16) = [ bx,   by,   bz, b2]   (per-column target point, b2=|b|^2)
//   C       = 0 (loop-invariant)
//   D       = |b_j|^2 - 2 a_i.b_j
// The row-constant |a_i|^2 is added once AFTER the j-loop min reduction
// (min_j(D + a2) == a2 + min_j D), so the hot loop has no C-splat movs
// and no per-tile adds: just load, 2 cndmask, 1 WMMA, min3 accumulation.
//
// min over the other direction comes from a second launch with A/B swapped,
// keeping all min state wave-local (no cross-wave atomics except one
// atomicAdd of the partial loss per wave).

typedef __attribute__((ext_vector_type(2))) float v2f;
typedef __attribute__((ext_vector_type(8))) float v8f;

#define BATCH 4
#define NPTS 8192
#define STRIPS_PER_BATCH (NPTS / 16)      // 512 row-strips per batch
#define WAVES_PER_BLOCK 8                 // 256 threads = 8 wave32
#define BLOCK_THREADS (WAVES_PER_BLOCK * 32)

// ds_swizzle group-of-32 xor patterns: offset[14:10]=xor, [9:5]=or, [4:0]=and
#define SWZ_X1  0x041F
#define SWZ_X2  0x081F
#define SWZ_X4  0x101F
#define SWZ_X8  0x201F
#define SWZ_X16 0x401F

template <int PAT>
__device__ __forceinline__ float xor_min(float x) {
  int o = __builtin_amdgcn_ds_swizzle(__float_as_int(x), PAT);
  return fminf(x, __int_as_float(o));
}

__global__ void recon_init_out(float* out) { out[0] = 0.0f; }

__global__ __launch_bounds__(BLOCK_THREADS)
void recon_chamfer_dir(const float* __restrict__ A,   // [BATCH, NPTS, 3] query points
                       const float* __restrict__ Bp,  // [BATCH, NPTS, 3] target points
                       float* __restrict__ out,       // scalar accumulator
                       float scale) {
  const int lane    = threadIdx.x & 31;
  const int wave    = threadIdx.x >> 5;
  const int strip   = blockIdx.x * WAVES_PER_BLOCK + wave;
  const int batch   = strip / STRIPS_PER_BATCH;
  const int rowbase = (strip % STRIPS_PER_BATCH) * 16;
  const int lane16  = lane & 15;
  const bool hi     = lane >= 16;

  // ---- A-matrix operand (16x4 fp32, ISA 7.12.2 layout) ----
  // lanes 0-15 : VGPR0=K0, VGPR1=K1 for row M=lane
  // lanes 16-31: VGPR0=K2, VGPR1=K3 for row M=lane-16
  const float* ap = A + ((size_t)batch * NPTS + rowbase + lane16) * 3;
  const float ax = ap[0], ay = ap[1], az = ap[2];
  const float a2 = ax * ax + ay * ay + az * az;   // added after the reduction
  v2f aReg;
  aReg.x = hi ? (-2.0f * az) : (-2.0f * ax);      // K2 / K0
  aReg.y = hi ? 1.0f         : (-2.0f * ay);      // K3 / K1 (K3=1 multiplies b2 in B)

  // Running row-minima of D = b2 - 2 a.b  (C/D layout: row = r + 8*(lane/16))
  v8f rmin;
#pragma unroll
  for (int r = 0; r < 8; ++r) rmin[r] = INFINITY;

  const v8f czero = {};                           // loop-invariant zero accumulator

  const float* bbase = Bp + (size_t)batch * NPTS * 3;

#pragma unroll 4
  for (int jt = 0; jt < NPTS; jt += 16) {
    // Each lane loads the target point of its column (N = lane%16).
    const float* bp = bbase + (size_t)(jt + lane16) * 3;
    const float bx = bp[0], by = bp[1], bz = bp[2];
    const float b2 = bx * bx + by * by + bz * bz;

    // ---- B-matrix operand (4x16 fp32): K rows striped across lane halves ----
    v2f bReg;
    bReg.x = hi ? bz : bx;                        // K2 / K0
    bReg.y = hi ? b2 : by;                        // K3 / K1 (b2 rides the K=3 row)

    // D = |b_j|^2 - 2 a_i.b_j
    v8f d = __builtin_amdgcn_wmma_f32_16x16x4_f32(
        false, aReg, false, bReg, (short)0, czero, false, false);

#pragma unroll
    for (int r = 0; r < 8; ++r) rmin[r] = fminf(rmin[r], d[r]);
  }

  // ---- Row-min reduction across the 16 lanes of each half (ds_swizzle xor) ----
  float s = 0.0f;
#pragma unroll
  for (int r = 0; r < 8; ++r) {
    float m = rmin[r];
    m = xor_min<SWZ_X1>(m);
    m = xor_min<SWZ_X2>(m);
    m = xor_min<SWZ_X4>(m);
    m = xor_min<SWZ_X8>(m);
    // add the row-constant |a_i|^2 for row (rowbase + r + 8*hi):
    // that lane loaded it as `a2` in lane (r) for the low half, lane (24+r) high.
    float a2r = __shfl(a2, hi ? (24 + r) : r, 32);
    s += sqrtf(fmaxf(m + a2r, 0.0f));
  }
  // low half summed rows rowbase..+7, high half rows rowbase+8..+15
  {
    int o = __builtin_amdgcn_ds_swizzle(__float_as_int(s), SWZ_X16);
    s += __int_as_float(o);
  }

  if (lane == 0) atomicAdd(out, s * scale);
}

extern "C" void kernel_launch(void* const* d_in, const int* in_sizes, int n_in,
                              void* d_out, int out_size, void* d_ws, size_t ws_size,
                              hipStream_t stream) {
  (void)in_sizes; (void)n_in; (void)out_size; (void)d_ws; (void)ws_size;
  const float* coords = (const float*)d_in[0];  // coords_transp [4,8192,3] f32
  const float* pc2    = (const float*)d_in[1];  // point_cloud_2 [4,8192,3] f32
  float* out = (float*)d_out;

  recon_init_out<<<1, 1, 0, stream>>>(out);

  const float scale = 1.0f / (float)(BATCH * NPTS);
  dim3 grid(BATCH * STRIPS_PER_BATCH / WAVES_PER_BLOCK);  // 256 blocks

  // Pass 1: for each coords point, nearest pc2 point  (min over dist rows)
  recon_chamfer_dir<<<grid, BLOCK_THREADS, 0, stream>>>(coords, pc2, out, scale);
  // Pass 2: symmetric direction, A/B swapped          (min over dist cols)
  recon_chamfer_dir<<<grid, BLOCK_THREADS, 0, stream>>>(pc2, coords, out, scale);
}